// VirtualPointAggerator_70970039599812
// MI455X (gfx1250) — compile-verified
//
#include <hip/hip_runtime.h>
#include <hip/hip_bf16.h>

typedef float v2f  __attribute__((ext_vector_type(2)));
typedef float v4f  __attribute__((ext_vector_type(4)));
typedef float v8f  __attribute__((ext_vector_type(8)));

#define KPTS    27
#define CCH     32
#define WSTRIDE 33                    // 32 weights + 1 pad -> no LDS bank conflicts
#define WAVE_LDS (16 * WSTRIDE + 16)  // weight tile + per-point exp factor, per wave

// exp(-x / (2*0.3^2 + 1e-9)) == exp2(x * NEG_C)
#define NEG_C ((float)(-(1.0 / (2.0 * 0.3 * 0.3 + 1e-9)) * 1.4426950408889634))

__global__ __launch_bounds__(256)
void vpa_kernel(const float* __restrict__ features,
                const float* __restrict__ rel,
                const float* __restrict__ kp,
                float* __restrict__ out,
                int N)
{
    __shared__ float lds[8 * WAVE_LDS];

    const int tid  = threadIdx.x;
    const int wave = tid >> 5;
    const int lane = tid & 31;
    const int m16  = lane & 15;    // col index (k) for B/D, row for A
    const int hi   = lane >> 4;    // lane-half selects K2/K3 of A, rows 8..15 of D

    float* wbase = &lds[wave * WAVE_LDS];
    float* efac  = wbase + 16 * WSTRIDE;      // exp2(|rp_m|^2 * NEG_C), per point

    const int numTiles = (N + 15) >> 4;
    const int tile = blockIdx.x * 8 + wave;   // wave-uniform
    if (tile >= numTiles) return;             // EXEC stays all-1s for live waves
    const int base = tile << 4;

    // ---------------- B matrices: kernel-point constants (4 x 16, K x N) ----
    // lanes 0-15: VGPR0=K0 row (-2*kx), VGPR1=K1 row (-2*ky)
    // lanes16-31: VGPR0=K2 row (-2*kz), VGPR1=K3 row (|kp|^2)
    const int k1 = m16;
    float k1x = kp[3*k1+0], k1y = kp[3*k1+1], k1z = kp[3*k1+2];
    v2f B1;
    B1.x = hi ? (-2.0f * k1z) : (-2.0f * k1x);
    B1.y = hi ? (k1x*k1x + k1y*k1y + k1z*k1z) : (-2.0f * k1y);

    const int k2 = 16 + m16;
    float k2x = 0.f, k2y = 0.f, k2z = 0.f;
    if (k2 < KPTS) { k2x = kp[3*k2+0]; k2y = kp[3*k2+1]; k2z = kp[3*k2+2]; }
    v2f B2;
    B2.x = hi ? (-2.0f * k2z) : (-2.0f * k2x);
    B2.y = hi ? (k2x*k2x + k2y*k2y + k2z*k2z) : (-2.0f * k2y);

    // ---------------- A matrix: 16 points x (rx, ry, rz, 1)  (16 x 4) -------
    int pA = base + m16; if (pA >= N) pA = N - 1;   // clamp; tail stores guarded
    float rx = rel[3*pA+0], ry = rel[3*pA+1], rz = rel[3*pA+2];
    v2f A;
    A.x = hi ? rz   : rx;
    A.y = hi ? 1.0f : ry;

    // per-point factor exp2(|rp|^2 * NEG_C): lanes 0-15 own point m16
    const float rp2 = rx*rx + ry*ry + rz*rz;
    if (!hi) efac[m16] = __builtin_amdgcn_exp2f(rp2 * NEG_C);

    // D[m][k] = |kp_k|^2 - 2 * rp_m . kp_k     (two 16x16 tiles of k)
    v8f Cz = {0.f,0.f,0.f,0.f,0.f,0.f,0.f,0.f};
    v8f D1 = __builtin_amdgcn_wmma_f32_16x16x4_f32(false, A, false, B1,
                                                   (short)0, Cz, false, false);
    v8f D2 = __builtin_amdgcn_wmma_f32_16x16x4_f32(false, A, false, B2,
                                                   (short)0, Cz, false, false);

    // ---------------- partial weights: exp2 straight off the accumulators ---
    // w[m][k] = efac[m] * exp2(D[m][k] * NEG_C); store only the second factor
    #pragma unroll
    for (int j = 0; j < 8; ++j) {
        const int m = j + 8 * hi;             // D layout: VGPR j, lane-half -> row
        const float w1 = __builtin_amdgcn_exp2f(D1[j] * NEG_C);
        const float w2 = __builtin_amdgcn_exp2f(D2[j] * NEG_C);
        wbase[m * WSTRIDE + m16]      = w1;   // k = m16
        wbase[m * WSTRIDE + 16 + m16] = w2;   // k = 16 + m16 (pad cols unused)
    }

    // ---------------- broadcast multiply + streaming stores -----------------
    // lane -> (channel quad cg, k-row offset kq); one b128 store = 512B/wave
    const int cg = lane & 7;                  // float4 index within 32 channels
    const int kq = lane >> 3;                 // 0..3
    for (int m = 0; m < 16; ++m) {
        const int p = base + m;
        if (p >= N) break;                    // wave-uniform
        const float em = efac[m];             // broadcast LDS read, 1 per point
        v4f f4 = *(const v4f*)(features + (size_t)p * CCH + cg * 4);
        f4 = f4 * em;                         // fold per-point exp factor once
        const size_t orow = (size_t)p * KPTS * CCH;
        #pragma unroll
        for (int kb = 0; kb < 28; kb += 4) {
            const int k = kb + kq;
            const float w = wbase[m * WSTRIDE + k];   // LDS broadcast (4 values)
            if (k < KPTS) {
                v4f o = f4 * w;
                __builtin_nontemporal_store(o,
                    (v4f*)(out + orow + (size_t)k * CCH + cg * 4));
            }
        }
    }
}

extern "C" void kernel_launch(void* const* d_in, const int* in_sizes, int n_in,
                              void* d_out, int out_size, void* d_ws, size_t ws_size,
                              hipStream_t stream)
{
    const float* features = (const float*)d_in[0];  // [N, 32]
    const float* rel      = (const float*)d_in[1];  // [N, 3]
    const float* kp       = (const float*)d_in[2];  // [27, 3]
    float*       out      = (float*)d_out;          // [N, 27, 32]

    const int N        = in_sizes[0] / CCH;
    const int numTiles = (N + 15) / 16;
    const int blocks   = (numTiles + 7) / 8;        // 8 waves/block, 1 tile/wave

    vpa_kernel<<<blocks, 256, 0, stream>>>(features, rel, kp, out, N);
}